// MeanFieldCRF_10453950398845
// MI455X (gfx1250) — compile-verified
//
#include <hip/hip_runtime.h>
#include <hip/hip_bf16.h>
#include <math.h>

// ---------------------------------------------------------------------------
// MeanFieldCRF on gfx1250 (MI455X).
// Message pass expressed as banded GEMM -> V_WMMA_F32_16X16X4_F32 (full f32).
// Filter (smoothing + appearance) precomputed once, stored directly in WMMA
// B-fragment order so the hot loop is: coalesced b64 load + 2 LDS reads + wmma.
// ---------------------------------------------------------------------------

#define B_    2
#define C_    6
#define CH_   3
#define H_    192
#define W_    192
#define FS_   11
#define PAD_  5
#define NITER 5

#define STRIPS   (W_ / 16)          // 12 strips per image row
#define LWP      (W_ + 12)          // padded LDS width: cols -5 .. W_+6  (204)
#define KCH      7                  // K chunks of 4 covering K = 28
#define FRAG_PER_STRIP (FS_ * KCH * 32 * 2)   // 11*7*64 = 4928 floats
#define FRAG_FLOATS ((size_t)B_ * H_ * STRIPS * FRAG_PER_STRIP)
#define PIX      (H_ * W_)          // 36864
#define NPIX     (B_ * PIX)         // 73728

typedef float v2f __attribute__((ext_vector_type(2)));
typedef float v8f __attribute__((ext_vector_type(8)));

// ---------------------------------------------------------------------------
// Kernel 1: build total_filter, scattered directly into WMMA B-fragment order.
//   frag[(b,h,strip)][i][kc][lane][t] = B[k = 4*kc + 2*(lane>>4) + t, n = lane&15]
//   B[k,n] = tf(w0+n, i, j=k-n) inside the band (0<=j<=10), else 0.
// Block = (h,b), 384 threads = 12 waves, wave s handles strip s.
// ---------------------------------------------------------------------------
__global__ __launch_bounds__(384)
void crf_filter_kernel(const float* __restrict__ feat,
                       const float* __restrict__ spacings,
                       const float* __restrict__ swp,
                       const float* __restrict__ awp,
                       const float* __restrict__ isbp,
                       const float* __restrict__ iabp,
                       float* __restrict__ frag) {
    const int h = blockIdx.x;
    const int b = blockIdx.y;
    const int tid = threadIdx.x;

    __shared__ float fs[CH_][FS_][LWP];   // zero-padded feature halo tile
    __shared__ float sfl[FS_ * FS_];      // per-batch smoothing filter

    for (int idx = tid; idx < CH_ * FS_ * LWP; idx += 384) {
        const int ch = idx / (FS_ * LWP);
        const int rem = idx % (FS_ * LWP);
        const int rl = rem / LWP;
        const int wp = rem % LWP;
        const int r = h + rl - PAD_;
        const int w = wp - PAD_;
        float v = 0.0f;
        if (r >= 0 && r < H_ && w >= 0 && w < W_)
            v = feat[(((size_t)b * CH_ + ch) * H_ + r) * W_ + w];
        fs[ch][rl][wp] = v;
    }
    if (tid < FS_ * FS_) {
        const int i = tid / FS_, j = tid % FS_;
        const float ibw = *isbp;
        const float di = (float)(i - PAD_) * spacings[b * 2 + 0] * ibw;
        const float dj = (float)(j - PAD_) * spacings[b * 2 + 1] * ibw;
        float v = __expf(-0.5f * (di * di + dj * dj));
        if (i == PAD_ && j == PAD_) v = 0.0f;   // zeroed center
        sfl[tid] = v;
    }
    __syncthreads();

    const float swv = *swp, awv = *awp, iab = *iabp;
    const float iab2h = 0.5f * iab * iab;

    const int lane = tid & 31;
    const int s = tid >> 5;                 // strip = wave id
    const int n = lane & 15;                // pixel-in-strip (B column)
    const int hi = lane >> 4;               // K half select
    const int w = s * 16 + n;
    const int wpc = w + PAD_;

    const float fc0 = fs[0][PAD_][wpc];
    const float fc1 = fs[1][PAD_][wpc];
    const float fc2 = fs[2][PAD_][wpc];

    float* fb = frag + (((size_t)(b * H_ + h) * STRIPS + s) * FRAG_PER_STRIP);

    for (int i = 0; i < FS_; ++i) {
        for (int kc = 0; kc < KCH; ++kc) {
            #pragma unroll
            for (int t = 0; t < 2; ++t) {
                const int k = 4 * kc + 2 * hi + t;
                const int j = k - n;
                float val = 0.0f;
                if (j >= 0 && j <= 10) {
                    const int wpp = w + j;          // padded col of patch
                    const float d0 = fc0 - fs[0][i][wpp];
                    const float d1 = fc1 - fs[1][i][wpp];
                    const float d2 = fc2 - fs[2][i][wpp];
                    const float af = __expf(-iab2h * (d0 * d0 + d1 * d1 + d2 * d2));
                    val = sfl[i * FS_ + j] * (swv + awv * af);
                }
                fb[((i * KCH + kc) * 32 + lane) * 2 + t] = val;
            }
        }
    }
}

// ---------------------------------------------------------------------------
// Kernel 2: softmax over the 6 classes (per pixel).
// ---------------------------------------------------------------------------
__global__ __launch_bounds__(256)
void crf_softmax_kernel(const float* __restrict__ x, float* __restrict__ q) {
    const int p = blockIdx.x * 256 + threadIdx.x;
    if (p >= NPIX) return;
    const int b = p / PIX;
    const int pix = p % PIX;
    const size_t base = (size_t)b * C_ * PIX + pix;
    float v[C_];
    float m = -3.4e38f;
    #pragma unroll
    for (int c = 0; c < C_; ++c) { v[c] = x[base + (size_t)c * PIX]; m = fmaxf(m, v[c]); }
    float ssum = 0.0f;
    #pragma unroll
    for (int c = 0; c < C_; ++c) { v[c] = __expf(v[c] - m); ssum += v[c]; }
    const float inv = 1.0f / ssum;
    #pragma unroll
    for (int c = 0; c < C_; ++c) q[base + (size_t)c * PIX] = v[c] * inv;
}

// ---------------------------------------------------------------------------
// Kernel 3: message pass + update, banded GEMM on V_WMMA_F32_16X16X4_F32.
//   D(16ch x 16pix) += A(q rows, 16x4) x B(banded filter fragments, 4x16),
//   accumulated over 11 kernel rows x 7 K-chunks. x_out = x0 + msg
//   (compat = -I  =>  x = unary - msg @ (-I) = unary + msg).
// Block = (h,b), 384 threads = 12 waves, wave s computes strip s.
// ---------------------------------------------------------------------------
__global__ __launch_bounds__(384)
void crf_msg_kernel(const float* __restrict__ q,
                    const float* __restrict__ frag,
                    const float* __restrict__ x0,
                    float* __restrict__ xo) {
    const int h = blockIdx.x;
    const int b = blockIdx.y;
    const int tid = threadIdx.x;

    __shared__ float qs[C_][FS_][LWP];    // zero-padded q halo tile (54 KB)

    for (int idx = tid; idx < C_ * FS_ * LWP; idx += 384) {
        const int c = idx / (FS_ * LWP);
        const int rem = idx % (FS_ * LWP);
        const int rl = rem / LWP;
        const int wp = rem % LWP;
        const int r = h + rl - PAD_;
        const int w = wp - PAD_;
        float v = 0.0f;
        if (r >= 0 && r < H_ && w >= 0 && w < W_)
            v = q[(((size_t)b * C_ + c) * H_ + r) * W_ + w];
        qs[c][rl][wp] = v;
    }
    __syncthreads();

    const int lane = tid & 31;
    const int s = tid >> 5;
    const int n = lane & 15;                 // A row (class) / B col (pixel)
    const int hi = lane >> 4;                // K half select
    const int cA = (n < C_) ? n : 0;         // clamp: rows 6..15 of D unused
    const int w0 = s * 16;

    const float* fb = frag + (((size_t)(b * H_ + h) * STRIPS + s) * FRAG_PER_STRIP);

    v8f acc = {0.f, 0.f, 0.f, 0.f, 0.f, 0.f, 0.f, 0.f};

    for (int i = 0; i < FS_; ++i) {
        #pragma unroll
        for (int kc = 0; kc < KCH; ++kc) {
            // B fragment: precomputed, coalesced 8B per lane (L2-resident)
            const float2 bf = *(const float2*)(fb + ((i * KCH + kc) * 32 + lane) * 2);
            // A fragment: q row segment from LDS; padded col index = w0 + k
            const int kb = 4 * kc + 2 * hi;
            v2f a;  a.x = qs[cA][i][w0 + kb];
                    a.y = qs[cA][i][w0 + kb + 1];
            v2f bv; bv.x = bf.x; bv.y = bf.y;
            acc = __builtin_amdgcn_wmma_f32_16x16x4_f32(
                      false, a, false, bv, (short)0, acc, false, false);
        }
    }

    // D layout: lane<16 holds rows m=0..7 (VGPR r -> class r); classes 0..5 valid.
    if (lane < 16) {
        #pragma unroll
        for (int r = 0; r < C_; ++r) {
            const size_t o = (((size_t)b * C_ + r) * H_ + h) * W_ + w0 + n;
            xo[o] = x0[o] + acc[r];
        }
    }
}

// ---------------------------------------------------------------------------
// Kernel 4: log_softmax in place.
// ---------------------------------------------------------------------------
__global__ __launch_bounds__(256)
void crf_logsoftmax_kernel(float* __restrict__ x) {
    const int p = blockIdx.x * 256 + threadIdx.x;
    if (p >= NPIX) return;
    const int b = p / PIX;
    const int pix = p % PIX;
    const size_t base = (size_t)b * C_ * PIX + pix;
    float v[C_];
    float m = -3.4e38f;
    #pragma unroll
    for (int c = 0; c < C_; ++c) { v[c] = x[base + (size_t)c * PIX]; m = fmaxf(m, v[c]); }
    float ssum = 0.0f;
    #pragma unroll
    for (int c = 0; c < C_; ++c) ssum += __expf(v[c] - m);
    const float l = m + __logf(ssum);
    #pragma unroll
    for (int c = 0; c < C_; ++c) x[base + (size_t)c * PIX] = v[c] - l;
}

// ---------------------------------------------------------------------------
extern "C" void kernel_launch(void* const* d_in, const int* in_sizes, int n_in,
                              void* d_out, int out_size, void* d_ws, size_t ws_size,
                              hipStream_t stream) {
    (void)in_sizes; (void)n_in; (void)out_size; (void)ws_size;

    const float* x0   = (const float*)d_in[0];   // (2,6,192,192)
    const float* feat = (const float*)d_in[1];   // (2,3,192,192)
    const float* sp   = (const float*)d_in[2];   // (2,2)
    const float* sw   = (const float*)d_in[3];
    const float* aw   = (const float*)d_in[4];
    const float* isb  = (const float*)d_in[5];
    const float* iab  = (const float*)d_in[6];
    float* out  = (float*)d_out;                 // x iterates here; final lsm in place
    float* frag = (float*)d_ws;                  // 22,708,224 floats (~86.6 MiB)
    float* q    = frag + FRAG_FLOATS;            // 442,368 floats

    const dim3 gridR(H_, B_);

    crf_filter_kernel<<<gridR, 384, 0, stream>>>(feat, sp, sw, aw, isb, iab, frag);

    for (int it = 0; it < NITER; ++it) {
        const float* xin = (it == 0) ? x0 : out;
        crf_softmax_kernel<<<NPIX / 256, 256, 0, stream>>>(xin, q);
        crf_msg_kernel<<<gridR, 384, 0, stream>>>(q, frag, x0, out);
    }

    crf_logsoftmax_kernel<<<NPIX / 256, 256, 0, stream>>>(out);
}